// CustomLoss_24567212933466
// MI455X (gfx1250) — compile-verified
//
#include <hip/hip_runtime.h>
#include <hip/hip_bf16.h>
#include <stdint.h>

#define TILE_F   1024   // floats per tile = 256 threads * float4
#define NTHREADS 256
#define DEPTH    4      // LDS staging buffers; up to DEPTH-1 tiles in flight

__device__ __forceinline__ void online_upd(float v, int gi, float& m, float& s, int& bi) {
    if (v > m) {
        // rescale old sum to new max, absorb this element (exp(0)=1)
        s = s * __expf(m - v) + 1.0f;
        m = v;
        bi = gi;
    } else {
        s += __expf(v - m);
    }
}

__global__ __launch_bounds__(NTHREADS) void row_softmax_stats(
    const float* __restrict__ logits, const int* __restrict__ tgt,
    float* __restrict__ nll_out, int* __restrict__ pred_out, int V) {

    const int row = blockIdx.x;
    const int tid = threadIdx.x;
    const float* rowp = logits + (size_t)row * (size_t)V;

    __shared__ float4 lbuf[DEPTH][NTHREADS];
    __shared__ float rm[8], rs[8];
    __shared__ int   ri[8];

    // LDS byte offset of lbuf base (flat shared pointer low 32 bits == LDS offset)
    const uint32_t lds_base = (uint32_t)(uintptr_t)(&lbuf[0][0]);

    const int nt = (V + TILE_F - 1) / TILE_F;

    float m = -INFINITY;
    float s = 0.0f;
    int   bi = 0x7fffffff;

    // Issue async global->LDS b128 for tile t (per-lane 16B, EXEC-masked on tail)
    auto issue = [&](int t) {
        int idx = t * TILE_F + tid * 4;
        if (idx < V) {
            uint32_t loff = lds_base + (uint32_t)(((t & (DEPTH - 1)) * NTHREADS + tid) * 16);
            uint32_t goff = (uint32_t)idx * 4u;   // byte offset from row base (SADDR)
            asm volatile("global_load_async_to_lds_b128 %0, %1, %2"
                         :: "v"(loff), "v"(goff), "s"(rowp)
                         : "memory");
        }
    };

    // Software pipeline, depth 4: keep up to 3 tiles in flight while consuming one.
    for (int t = 0; t < DEPTH - 1 && t < nt; ++t) issue(t);

    for (int t = 0; t < nt; ++t) {
        if (t + DEPTH - 1 < nt) issue(t + DEPTH - 1);

        // Async loads complete in order: wait until at most (#tiles newer than t)
        // remain outstanding => tile t has landed in LDS.
        if      (t + 3 < nt) asm volatile("s_wait_asynccnt 0x3" ::: "memory");
        else if (t + 2 < nt) asm volatile("s_wait_asynccnt 0x2" ::: "memory");
        else if (t + 1 < nt) asm volatile("s_wait_asynccnt 0x1" ::: "memory");
        else                 asm volatile("s_wait_asynccnt 0x0" ::: "memory");

        int idx = t * TILE_F + tid * 4;
        if (idx + 3 < V) {
            float4 x = lbuf[t & (DEPTH - 1)][tid];
            online_upd(x.x, idx + 0, m, s, bi);
            online_upd(x.y, idx + 1, m, s, bi);
            online_upd(x.z, idx + 2, m, s, bi);
            online_upd(x.w, idx + 3, m, s, bi);
        } else if (idx < V) {
            const float* lp = (const float*)&lbuf[t & (DEPTH - 1)][tid];
            #pragma unroll
            for (int k = 0; k < 4; ++k)
                if (idx + k < V) online_upd(lp[k], idx + k, m, s, bi);
        }
    }

    // wave32 reduction: combine (m, s) with rescaling; argmax tie -> smaller index
    #pragma unroll
    for (int off = 16; off > 0; off >>= 1) {
        float om = __shfl_xor(m, off, 32);
        float os = __shfl_xor(s, off, 32);
        int   ob = __shfl_xor(bi, off, 32);
        float nm = fmaxf(m, om);
        float ns = s * __expf(m - nm) + os * __expf(om - nm);
        bi = (om > m) ? ob : ((m > om) ? bi : (ob < bi ? ob : bi));
        m = nm; s = ns;
    }

    const int lane = tid & 31, wid = tid >> 5;
    if (lane == 0) { rm[wid] = m; rs[wid] = s; ri[wid] = bi; }
    __syncthreads();

    if (tid == 0) {
        float M = rm[0], S = rs[0];
        int   I = ri[0];
        const int nw = blockDim.x >> 5;
        for (int w = 1; w < nw; ++w) {
            float om = rm[w], os = rs[w];
            int   ob = ri[w];
            float nm = fmaxf(M, om);
            float ns = S * __expf(M - nm) + os * __expf(om - nm);
            I = (om > M) ? ob : ((M > om) ? I : (ob < I ? ob : I));
            M = nm; S = ns;
        }
        int tv = tgt[row];
        float xt = rowp[tv];
        nll_out[row]  = M + logf(S) - xt;  // -(x_t - max - log(sum))
        pred_out[row] = I;
    }
}

__global__ __launch_bounds__(NTHREADS) void finalize_loss(
    const float* __restrict__ nll, const int* __restrict__ pred,
    const float* __restrict__ sizes, float* __restrict__ out,
    int BT, int T, int V, float invBT) {

    __shared__ float ssum[NTHREADS];
    __shared__ int   cnt[16];

    const int tid = threadIdx.x;
    if (tid < 16) cnt[tid] = 0;
    __syncthreads();

    float acc = 0.0f;
    for (int r = tid; r < BT; r += NTHREADS) {
        acc += nll[r];
        int b = r / T;
        int p = pred[r];
        float e = sizes[(size_t)b * (size_t)V + (size_t)p];
        if (e > 0.0f) atomicAdd(&cnt[b], 1);
    }
    ssum[tid] = acc;
    __syncthreads();

    #pragma unroll
    for (int off = NTHREADS / 2; off > 0; off >>= 1) {
        if (tid < off) ssum[tid] += ssum[tid + off];
        __syncthreads();
    }

    if (tid == 0) {
        // penalty = sum_t m_t * (#ones strictly before t) = K*(K-1)/2 per batch
        const int B = BT / T;
        float pen = 0.0f;
        for (int b = 0; b < B; ++b) {
            float k = (float)cnt[b];
            pen += 0.5f * k * (k - 1.0f);
        }
        out[0] = ssum[0] * invBT + pen;   // ALPHA = 1
    }
}

extern "C" void kernel_launch(void* const* d_in, const int* in_sizes, int n_in,
                              void* d_out, int out_size, void* d_ws, size_t ws_size,
                              hipStream_t stream) {
    const float* logits = (const float*)d_in[0];
    const int*   tgt    = (const int*)d_in[1];
    const float* sizes  = (const float*)d_in[2];
    float*       out    = (float*)d_out;

    const long long n0 = (long long)in_sizes[0];   // B*T*V
    const int BT = in_sizes[1];                    // B*T
    const int BV = in_sizes[2];                    // B*V
    const int V  = (int)(n0 / (long long)BT);
    const int B  = BV / V;
    const int T  = BT / B;

    float* nll  = (float*)d_ws;          // BT floats
    int*   pred = (int*)(nll + BT);      // BT ints

    row_softmax_stats<<<BT, NTHREADS, 0, stream>>>(logits, tgt, nll, pred, V);
    finalize_loss<<<1, NTHREADS, 0, stream>>>(nll, pred, sizes, out, BT, T, V,
                                              1.0f / (float)BT);
}